// HypAgg_54649163874379
// MI455X (gfx1250) — compile-verified
//
#include <hip/hip_runtime.h>

// ---------------------------------------------------------------------------
// HypAgg (Poincare-ball graph attention aggregation) for MI455X / gfx1250.
// B=4, N=256, D=128, C=1. All-pairs edges with edge_mask = 1 - I.
//
// Math refactor: logmap(p1,p2) = s*(a*p1 + b*p2) with scalars a,b,s from
// (|p1|^2, |p2|^2, <p1,p2>).  Attention logits and the neighbor aggregation
// therefore reduce to scalar pair-math plus dense GEMMs:
//   G = x x^T (WMMA) -> per-pair scalars -> M (VALU+LDS) ->
//   support = M x (WMMA) -> 2-layer MLP (WMMA, fused SiLU) -> expmap+proj.
// Avoids the 128MB [B,N,N,D] intermediate entirely (~4MB workspace).
// ---------------------------------------------------------------------------

typedef __attribute__((ext_vector_type(2))) float v2f;
typedef __attribute__((ext_vector_type(8))) float v8f;

#define BATCH 4
#define NNODE 256
#define DDIM  128
#define MIN_NORM 1e-15f
#define BALL_EPS 4e-3f

// ---- block-wide sum reduction (power-of-two nthr), result broadcast ----
__device__ __forceinline__ float block_reduce_sum(float v, float* sbuf, int tid, int nthr) {
  sbuf[tid] = v;
  __syncthreads();
  for (int s = nthr >> 1; s > 0; s >>= 1) {
    if (tid < s) sbuf[tid] += sbuf[tid + s];
    __syncthreads();
  }
  float r = sbuf[0];
  __syncthreads();
  return r;
}

// ---- generic fp32 WMMA 16x16 tile GEMM: A[MxK] row-major, B[KxN] row-major.
// ISA 7.12.2 layouts: A 16x4 -> 2 VGPRs: lanes 0-15 K={0,1}, lanes 16-31
// K={2,3}; B 4x16 -> 2 VGPRs mirrored; C/D: VGPR r holds rows r (lanes 0-15)
// and r+8 (lanes 16-31). Wave32, EXEC all-1s (uniform control flow).
__device__ __forceinline__ v8f wmma_tile_f32(const float* __restrict__ A, int lda,
                                             const float* __restrict__ B, int ldb,
                                             int i0, int j0, int K, int lane) {
  const int half = lane >> 4;
  const int l = lane & 15;
  const float* ar = A + (size_t)(i0 + l) * lda;
  v8f acc = {};
  for (int k0 = 0; k0 < K; k0 += 4) {
    const int ka = k0 + half * 2;
    v2f a;  a.x = ar[ka];                          a.y = ar[ka + 1];
    v2f bv; bv.x = B[(size_t)ka * ldb + j0 + l];   bv.y = B[(size_t)(ka + 1) * ldb + j0 + l];
    acc = __builtin_amdgcn_wmma_f32_16x16x4_f32(false, a, false, bv, (short)0, acc,
                                                false, false);
  }
  return acc;
}

// ---- K0: per-node stats: n2 = |x_i|^2, g = <x_i, Watt[0:D]> ----
__global__ void stats_kernel(const float* __restrict__ x, const float* __restrict__ Watt,
                             float* __restrict__ n2, float* __restrict__ g) {
  const int node = blockIdx.x;     // b*N + i
  const int d = threadIdx.x;       // 0..127
  __shared__ float sbuf[DDIM];
  const float v = x[(size_t)node * DDIM + d];
  const float w = Watt[d];         // only the first D rows matter (x_self == 0)
  const float s2 = block_reduce_sum(v * v, sbuf, d, DDIM);
  const float sg = block_reduce_sum(v * w, sbuf, d, DDIM);
  if (d == 0) { n2[node] = s2; g[node] = sg; }
}

// ---- K1: Gram matrix G[b] = x[b] x[b]^T via v_wmma_f32_16x16x4_f32 ----
__global__ void gram_kernel(const float* __restrict__ x, float* __restrict__ G) {
  const int i0 = blockIdx.x * 16, j0 = blockIdx.y * 16, b = blockIdx.z;
  const int lane = threadIdx.x, half = lane >> 4, l = lane & 15;
  const float* xb = x + (size_t)b * NNODE * DDIM;
  const float* arow = xb + (size_t)(i0 + l) * DDIM;
  const float* brow = xb + (size_t)(j0 + l) * DDIM;  // B = x^T: B[k,n] = x[n,k]
  v8f acc = {};
  for (int k0 = 0; k0 < DDIM; k0 += 4) {
    const int ka = k0 + half * 2;
    v2f a;  a.x = arow[ka];  a.y = arow[ka + 1];
    v2f bv; bv.x = brow[ka]; bv.y = brow[ka + 1];
    acc = __builtin_amdgcn_wmma_f32_16x16x4_f32(false, a, false, bv, (short)0, acc,
                                                false, false);
  }
  float* Gt = G + ((size_t)b * NNODE + i0) * NNODE + j0;
  for (int r = 0; r < 8; ++r) Gt[(size_t)(r + half * 8) * NNODE + l] = acc[r];
}

// ---- K2: per-pair scalar math -> aggregation matrix M[b] ----
// M[i,j] = w*b_coef (i!=j), M[i,i] = sum_j w*a_coef  (diag masked out anyway).
__global__ void pair_kernel(const float* __restrict__ G, const float* __restrict__ n2,
                            const float* __restrict__ g, const float* __restrict__ edge_mask,
                            const float* __restrict__ batt, float* __restrict__ M) {
  const int node = blockIdx.x;           // b*N + i
  const int b = node >> 8, i = node & 255;
  const int j = threadIdx.x;             // 0..255
  __shared__ float sbuf[NNODE];

  const float dot  = G[(size_t)node * NNODE + j];
  const float x2   = n2[node];
  const float y2   = n2[b * NNODE + j];
  const float gi   = g[node];
  const float gj   = g[b * NNODE + j];
  const float mask = edge_mask[(size_t)node * NNODE + j];

  // mobius_add(-p1, p2): sub = a*p1 + bc*p2
  const float tw  = 1.f - 2.f * dot + y2;
  const float den = fmaxf(1.f - 2.f * dot + x2 * y2, MIN_NORM);
  const float a   = -tw / den;
  const float bc  = (1.f - x2) / den;
  const float sn2 = a * a * x2 + 2.f * a * bc * dot + bc * bc * y2;
  const float sn  = fmaxf(sqrtf(fmaxf(sn2, 0.f)), MIN_NORM);
  const float lam = 2.f / fmaxf(1.f - x2, MIN_NORM);
  const float z   = fminf(sn, 1.f - 1e-7f);
  const float ath = 0.5f * logf((1.f + z) / (1.f - z));   // artanh
  const float s   = (2.f / lam) * ath / sn;               // logmap = s*sub
  // attention logit: <mask*x_local, Watt[0:D]> + batt
  const float logit = mask * (s * (a * gi + bc * gj)) + batt[0];
  const float att = 1.f / (1.f + expf(-logit));
  const float w = att * mask * s;

  if (j != i) M[(size_t)node * NNODE + j] = w * bc;
  const float c = block_reduce_sum(w * a, sbuf, j, NNODE);
  if (j == 0) M[(size_t)node * NNODE + i] = c;   // x_i coefficient on diagonal
}

// ---- K3: support[b] = M[b] @ x[b]  (WMMA) ----
__global__ void support_gemm(const float* __restrict__ M, const float* __restrict__ x,
                             float* __restrict__ S) {
  const int i0 = blockIdx.x * 16, d0 = blockIdx.y * 16, b = blockIdx.z;
  const int lane = threadIdx.x, half = lane >> 4, l = lane & 15;
  const float* Mb = M + (size_t)b * NNODE * NNODE;
  const float* xb = x + (size_t)b * NNODE * DDIM;
  v8f acc = wmma_tile_f32(Mb, NNODE, xb, DDIM, i0, d0, NNODE, lane);
  float* St = S + ((size_t)b * NNODE + i0) * DDIM + d0;
  for (int r = 0; r < 8; ++r) St[(size_t)(r + half * 8) * DDIM + l] = acc[r];
}

// ---- K4a: H1 = silu(S @ W1[D:2D,:] + b1)  (WMMA, fused epilogue) ----
__global__ void mlp1_gemm(const float* __restrict__ S, const float* __restrict__ W1,
                          const float* __restrict__ b1, float* __restrict__ H1) {
  const int i0 = blockIdx.x * 16, j0 = blockIdx.y * 16;
  const int lane = threadIdx.x, half = lane >> 4, l = lane & 15;
  const float* W1b = W1 + (size_t)DDIM * DDIM;   // rows D..2D-1 (x_self half is zero)
  v8f acc = wmma_tile_f32(S, DDIM, W1b, DDIM, i0, j0, DDIM, lane);
  const float bias = b1[j0 + l];
  float* Ht = H1 + (size_t)i0 * DDIM + j0;
  for (int r = 0; r < 8; ++r) {
    const float t = acc[r] + bias;
    Ht[(size_t)(r + half * 8) * DDIM + l] = t / (1.f + expf(-t));   // SiLU
  }
}

// ---- K4b: H = H1 @ W2 + b2  (WMMA, fused bias) ----
__global__ void mlp2_gemm(const float* __restrict__ H1, const float* __restrict__ W2,
                          const float* __restrict__ b2, float* __restrict__ H) {
  const int i0 = blockIdx.x * 16, j0 = blockIdx.y * 16;
  const int lane = threadIdx.x, half = lane >> 4, l = lane & 15;
  v8f acc = wmma_tile_f32(H1, DDIM, W2, DDIM, i0, j0, DDIM, lane);
  const float bias = b2[j0 + l];
  float* Ht = H + (size_t)i0 * DDIM + j0;
  for (int r = 0; r < 8; ++r)
    Ht[(size_t)(r + half * 8) * DDIM + l] = acc[r] + bias;
}

// ---- K5: out = proj(expmap(H, x)) ----
__global__ void expmap_kernel(const float* __restrict__ H, const float* __restrict__ x,
                              float* __restrict__ out) {
  const int node = blockIdx.x;
  const int d = threadIdx.x;        // 0..127
  __shared__ float sbuf[DDIM];
  const float u = H[(size_t)node * DDIM + d];
  const float p = x[(size_t)node * DDIM + d];
  const float su2 = block_reduce_sum(u * u, sbuf, d, DDIM);
  const float sp2 = block_reduce_sum(p * p, sbuf, d, DDIM);
  const float spu = block_reduce_sum(p * u, sbuf, d, DDIM);
  const float un  = fmaxf(sqrtf(su2), MIN_NORM);
  const float lam = 2.f / fmaxf(1.f - sp2, MIN_NORM);
  const float f   = tanhf(0.5f * lam * un) / un;   // second = f*u
  const float y2  = f * f * su2;
  const float xy  = f * spu;
  const float den = fmaxf(1.f + 2.f * xy + sp2 * y2, MIN_NORM);
  const float o = ((1.f + 2.f * xy + y2) * p + (1.f - sp2) * (f * u)) / den;
  const float so2 = block_reduce_sum(o * o, sbuf, d, DDIM);
  const float on = fmaxf(sqrtf(so2), MIN_NORM);
  const float maxn = 1.f - BALL_EPS;
  const float scale = (on > maxn) ? (maxn / on) : 1.f;
  out[(size_t)node * DDIM + d] = o * scale;
}

extern "C" void kernel_launch(void* const* d_in, const int* in_sizes, int n_in,
                              void* d_out, int out_size, void* d_ws, size_t ws_size,
                              hipStream_t stream) {
  // setup_inputs order: x, distances(unused), edges(unused: fixed all-pairs),
  // node_mask(unused by reference), edge_mask, Watt, batt, W1, b1, W2, b2
  const float* x         = (const float*)d_in[0];
  const float* edge_mask = (const float*)d_in[4];
  const float* Watt      = (const float*)d_in[5];
  const float* batt      = (const float*)d_in[6];
  const float* W1        = (const float*)d_in[7];
  const float* b1        = (const float*)d_in[8];
  const float* W2        = (const float*)d_in[9];
  const float* b2        = (const float*)d_in[10];
  float* out = (float*)d_out;

  // workspace layout (floats): G[262144] M[262144] n2[1024] g[1024]
  //                            S[131072] H1[131072] H[131072]  (~3.7 MB)
  float* ws = (float*)d_ws;
  float* G  = ws;
  float* M  = ws + 262144;
  float* n2 = ws + 524288;
  float* g  = ws + 525312;
  float* S  = ws + 526336;
  float* H1 = ws + 657408;
  float* H  = ws + 788480;

  stats_kernel<<<BATCH * NNODE, DDIM, 0, stream>>>(x, Watt, n2, g);
  gram_kernel<<<dim3(NNODE / 16, NNODE / 16, BATCH), 32, 0, stream>>>(x, G);
  pair_kernel<<<BATCH * NNODE, NNODE, 0, stream>>>(G, n2, g, edge_mask, batt, M);
  support_gemm<<<dim3(NNODE / 16, DDIM / 16, BATCH), 32, 0, stream>>>(M, x, S);
  mlp1_gemm<<<dim3(BATCH * NNODE / 16, DDIM / 16), 32, 0, stream>>>(S, W1, b1, H1);
  mlp2_gemm<<<dim3(BATCH * NNODE / 16, DDIM / 16), 32, 0, stream>>>(H1, W2, b2, H);
  expmap_kernel<<<BATCH * NNODE, DDIM, 0, stream>>>(H, x, out);
}